// HGPSLPool__52974126629105
// MI455X (gfx1250) — compile-verified
//
#include <hip/hip_runtime.h>
#include <hip/hip_bf16.h>
#include <math.h>

// Problem constants (from reference)
constexpr int N = 16000;
constexpr int E = 512000;
constexpr int C = 128;        // channels
constexpr int K = 12800;      // ceil(0.8 * N)
constexpr float SLOPE = 0.2f;
constexpr float LAMB  = 1.0f;

typedef __attribute__((ext_vector_type(2))) float v2f;
typedef __attribute__((ext_vector_type(8))) float v8f;

__device__ __forceinline__ float leaky(float v) { return v >= 0.f ? v : SLOPE * v; }

// ---------------- init: zero deg / edge-bucket counters ----------------
__global__ void k_zero(float* __restrict__ deg, int* __restrict__ cnt, int* __restrict__ fill) {
    int i = blockIdx.x * blockDim.x + threadIdx.x;
    if (i < N) deg[i] = 0.f;
    if (i < K) { cnt[i] = 0; fill[i] = 0; }
}

__global__ void k_deg(const int* __restrict__ row, float* __restrict__ deg) {
    int e = blockIdx.x * blockDim.x + threadIdx.x;
    if (e < E) atomicAdd(&deg[row[e]], 1.0f);
}

__global__ void k_dinv(const float* __restrict__ deg, float* __restrict__ dinv) {
    int i = blockIdx.x * blockDim.x + threadIdx.x;
    if (i < N) {
        float d = deg[i] + 1.0f;                 // +1 self loop
        dinv[i] = 1.0f / sqrtf(d);               // d >= 1 always
    }
}

// info[i][:] = dinv[i]^2 * x[i][:]   (self-loop term initializes info)
__global__ void k_info_self(const float* __restrict__ x, const float* __restrict__ dinv,
                            float* __restrict__ info) {
    int t = blockIdx.x * blockDim.x + threadIdx.x;   // N*32 threads, float4 each
    if (t >= N * (C / 4)) return;
    int i = t >> 5, q = t & 31;
    float d = dinv[i];
    float s = d * d;
    const float4 xv = *(const float4*)&x[(size_t)i * C + q * 4];
    float4 o; o.x = s * xv.x; o.y = s * xv.y; o.z = s * xv.z; o.w = s * xv.w;
    *(float4*)&info[(size_t)i * C + q * 4] = o;
}

// info[col][:] += dinv[row]*w*dinv[col] * x[row][:]
__global__ void k_info_edge(const float* __restrict__ x, const int* __restrict__ row,
                            const int* __restrict__ col, const float* __restrict__ ea,
                            const float* __restrict__ dinv, float* __restrict__ info) {
    long long t = (long long)blockIdx.x * blockDim.x + threadIdx.x;  // E*32 threads
    if (t >= (long long)E * (C / 4)) return;
    int e = (int)(t >> 5), q = (int)(t & 31);
    int r = row[e], c = col[e];
    float norm = dinv[r] * ea[e] * dinv[c];
    const float4 xv = *(const float4*)&x[(size_t)r * C + q * 4];
    float* dst = &info[(size_t)c * C + q * 4];
    atomicAdd(dst + 0, norm * xv.x);
    atomicAdd(dst + 1, norm * xv.y);
    atomicAdd(dst + 2, norm * xv.z);
    atomicAdd(dst + 3, norm * xv.w);
}

__global__ void k_score(const float* __restrict__ info, float* __restrict__ score) {
    int i = blockIdx.x * blockDim.x + threadIdx.x;
    if (i >= N) return;
    float s = 0.f;
    const float* p = &info[(size_t)i * C];
    for (int c = 0; c < C; ++c) s += fabsf(p[c]);
    score[i] = s;
}

// Stable descending rank (== jax.lax.top_k order: value desc, ties -> lower index)
__global__ void k_rank(const float* __restrict__ score, int* __restrict__ perm,
                       int* __restrict__ newidx) {
    int i = blockIdx.x * blockDim.x + threadIdx.x;
    if (i >= N) return;
    float si = score[i];
    int r = 0;
    for (int j = 0; j < N; ++j) {
        float sj = score[j];
        r += (sj > si) || (sj == si && j < i);
    }
    if (r < K) { perm[r] = i; newidx[i] = r; }
    else       { newidx[i] = -1; }
}

__global__ void k_gather_xk(const float* __restrict__ x, const int* __restrict__ perm,
                            float* __restrict__ xk) {
    int t = blockIdx.x * blockDim.x + threadIdx.x;   // K*32 threads
    if (t >= K * (C / 4)) return;
    int j = t >> 5, q = t & 31;
    int src = perm[j];
    *(float4*)&xk[(size_t)j * C + q * 4] = *(const float4*)&x[(size_t)src * C + q * 4];
}

// ---------------- s1/s2 via V_WMMA_F32_16X16X4_F32 ----------------
// S = xk (K x 128) @ B (128 x 16), B col0 = a1, col1 = a2, rest 0.
// One wave per 16-row tile; 32 accumulating WMMAs over the 128-deep K dim.
// Branchless fragment packing -> no EXEC churn between WMMAs.
__global__ __launch_bounds__(32) void k_s_wmma(const float* __restrict__ xk,
                                               const float* __restrict__ att,
                                               float* __restrict__ s1,
                                               float* __restrict__ s2) {
    const int tile = blockIdx.x;          // K/16 tiles
    const int lane = threadIdx.x;         // 0..31
    const int n    = lane & 15;
    const int half = lane >> 4;           // 0 or 1
#if __has_builtin(__builtin_amdgcn_wmma_f32_16x16x4_f32)
    v8f acc = {};
    const int   m_row   = tile * 16 + n;           // A fragment: m = lane & 15
    const float* arow   = xk + (size_t)m_row * C;
    const int   colBase = (n == 1) ? C : 0;        // b-col 0 -> a1, col 1 -> a2
    const float msk     = (n < 2) ? 1.f : 0.f;     // cols 2..15 of B are zero
    #pragma unroll 8
    for (int kc = 0; kc < C; kc += 4) {
        const int ka = kc + half * 2;     // lanes 0-15: K 0,1 ; lanes 16-31: K 2,3
        v2f a = *(const v2f*)&arow[ka];                    // 8B aligned (ka even)
        v2f b;
        b.x = msk * att[colBase + ka];
        b.y = msk * att[colBase + ka + 1];
        acc = __builtin_amdgcn_wmma_f32_16x16x4_f32(
            false, a, false, b, (short)0, acc, false, false);
    }
    // D(m,n): vgpr r = m & 7, lane = (m>=8 ? 16 : 0) + n  ->  m = half*8 + r
    if (n == 0) {
        #pragma unroll
        for (int r = 0; r < 8; ++r) s1[tile * 16 + half * 8 + r] = acc[r];
    } else if (n == 1) {
        #pragma unroll
        for (int r = 0; r < 8; ++r) s2[tile * 16 + half * 8 + r] = acc[r];
    }
#else
    // Scalar fallback (also what the host-side parse sees)
    if (lane < 16) {
        int r = tile * 16 + lane;
        float acc1 = 0.f, acc2 = 0.f;
        for (int c = 0; c < C; ++c) {
            float v = xk[(size_t)r * C + c];
            acc1 += v * att[c];
            acc2 += v * att[C + c];
        }
        s1[r] = acc1; s2[r] = acc2;
    }
#endif
}

// ---------------- CSR-bucket the valid edges by new row index ----------------
__global__ void k_count(const int* __restrict__ row, const int* __restrict__ col,
                        const int* __restrict__ newidx, int* __restrict__ cnt) {
    int e = blockIdx.x * blockDim.x + threadIdx.x;
    if (e >= E) return;
    int nr = newidx[row[e]];
    int nc = newidx[col[e]];
    if (nr >= 0 && nc >= 0) atomicAdd(&cnt[nr], 1);
}

// Single-workgroup exclusive scan of cnt[K] -> offs[K+1]
__global__ __launch_bounds__(1024) void k_scan(const int* __restrict__ cnt,
                                               int* __restrict__ offs) {
    __shared__ int part[1024];
    constexpr int CH = (K + 1023) / 1024;   // 13 per thread
    const int t = threadIdx.x;
    int s = 0;
    for (int i = 0; i < CH; ++i) {
        int idx = t * CH + i;
        if (idx < K) s += cnt[idx];
    }
    part[t] = s; __syncthreads();
    for (int off = 1; off < 1024; off <<= 1) {      // inclusive Hillis-Steele
        int add = (t >= off) ? part[t - off] : 0;
        __syncthreads();
        part[t] += add;
        __syncthreads();
    }
    int base = (t == 0) ? 0 : part[t - 1];
    for (int i = 0; i < CH; ++i) {
        int idx = t * CH + i;
        if (idx < K) { offs[idx] = base; base += cnt[idx]; }
    }
    if (t == 1023) offs[K] = part[1023];
}

__global__ void k_fill(const int* __restrict__ row, const int* __restrict__ col,
                       const float* __restrict__ ea, const int* __restrict__ newidx,
                       const int* __restrict__ offs, int* __restrict__ fill,
                       int* __restrict__ ecol, float* __restrict__ eval) {
    int e = blockIdx.x * blockDim.x + threadIdx.x;
    if (e >= E) return;
    int nr = newidx[row[e]];
    int nc = newidx[col[e]];
    if (nr >= 0 && nc >= 0) {
        int p = offs[nr] + atomicAdd(&fill[nr], 1);
        ecol[p] = nc;
        eval[p] = ea[e] * LAMB;
    }
}

// ---------------- fused: build row + edge scatter + softmax, one HBM write ----
__global__ __launch_bounds__(256) void k_row_fused(const float* __restrict__ s1,
                                                   const float* __restrict__ s2,
                                                   const int*   __restrict__ offs,
                                                   const int*   __restrict__ ecol,
                                                   const float* __restrict__ eval,
                                                   float* __restrict__ W) {
    __shared__ float buf[K];          // 51.2 KB row buffer (LDS, 320 KB/WGP)
    __shared__ float red[256];
    const int r = blockIdx.x;
    const int t = threadIdx.x;
    const int b0 = offs[r], b1 = offs[r + 1];
    __builtin_prefetch(&ecol[b0], 0, 0);            // global_prefetch_b8
    const float a = s1[r];
    for (int c = t; c < K; c += 256) buf[c] = leaky(a + s2[c]);
    __syncthreads();
    for (int i = b0 + t; i < b1; i += 256)          // LDS atomic scatter
        atomicAdd(&buf[ecol[i]], eval[i]);
    __syncthreads();
    // row max
    float m = -INFINITY;
    for (int c = t; c < K; c += 256) m = fmaxf(m, buf[c]);
    red[t] = m; __syncthreads();
    for (int s = 128; s > 0; s >>= 1) {
        if (t < s) red[t] = fmaxf(red[t], red[t + s]);
        __syncthreads();
    }
    m = red[0]; __syncthreads();
    // exp + sum
    float sum = 0.f;
    for (int c = t; c < K; c += 256) {
        float e = expf(buf[c] - m);
        buf[c] = e;
        sum += e;
    }
    red[t] = sum; __syncthreads();
    for (int s = 128; s > 0; s >>= 1) {
        if (t < s) red[t] += red[t + s];
        __syncthreads();
    }
    const float inv = 1.0f / red[0];
    const size_t base = (size_t)r * K;
    for (int c = t; c < K; c += 256) W[base + c] = buf[c] * inv;
}

__global__ void k_batch(float* __restrict__ batch) {
    int i = blockIdx.x * blockDim.x + threadIdx.x;
    if (i < K) batch[i] = 0.f;   // int32 zeros == float zeros bitwise
}

extern "C" void kernel_launch(void* const* d_in, const int* in_sizes, int n_in,
                              void* d_out, int out_size, void* d_ws, size_t ws_size,
                              hipStream_t stream) {
    const float* x    = (const float*)d_in[0];
    const int*   ei   = (const int*)d_in[1];
    const float* ea   = (const float*)d_in[2];
    const float* att  = (const float*)d_in[3];
    const int* row = ei;
    const int* col = ei + E;

    // workspace layout
    float* ws    = (float*)d_ws;
    float* deg   = ws;                    // N
    float* dinv  = ws + N;                // N
    float* score = ws + 2 * N;            // N
    float* info  = ws + 3 * N;            // N*C
    float* s1    = info + (size_t)N * C;  // K
    float* s2    = s1 + K;                // K
    float* eval  = s2 + K;                // E
    int*   perm   = (int*)(eval + E);     // K
    int*   newidx = perm + K;             // N
    int*   cnt    = newidx + N;           // K
    int*   fill   = cnt + K;              // K
    int*   offs   = fill + K;             // K+1
    int*   ecol   = offs + K + 1;         // E

    // output layout: xk | new_edge_attr | batch
    float* xk    = (float*)d_out;                       // K*C
    float* W     = xk + (size_t)K * C;                  // K*K
    float* batch = W + (size_t)K * K;                   // K

    const int B = 256;
    k_zero      <<<(N + B - 1) / B, B, 0, stream>>>(deg, cnt, fill);
    k_deg       <<<(E + B - 1) / B, B, 0, stream>>>(row, deg);
    k_dinv      <<<(N + B - 1) / B, B, 0, stream>>>(deg, dinv);
    k_info_self <<<(N * (C / 4) + B - 1) / B, B, 0, stream>>>(x, dinv, info);
    k_info_edge <<<(int)(((long long)E * (C / 4) + B - 1) / B), B, 0, stream>>>(x, row, col, ea, dinv, info);
    k_score     <<<(N + B - 1) / B, B, 0, stream>>>(info, score);
    k_rank      <<<(N + B - 1) / B, B, 0, stream>>>(score, perm, newidx);
    k_gather_xk <<<(K * (C / 4) + B - 1) / B, B, 0, stream>>>(x, perm, xk);
    k_s_wmma    <<<K / 16, 32, 0, stream>>>(xk, att, s1, s2);
    k_count     <<<(E + B - 1) / B, B, 0, stream>>>(row, col, newidx, cnt);
    k_scan      <<<1, 1024, 0, stream>>>(cnt, offs);
    k_fill      <<<(E + B - 1) / B, B, 0, stream>>>(row, col, ea, newidx, offs, fill, ecol, eval);
    k_row_fused <<<K, 256, 0, stream>>>(s1, s2, offs, ecol, eval, W);
    k_batch     <<<(K + B - 1) / B, B, 0, stream>>>(batch);
}